// T3A_9474697855065
// MI455X (gfx1250) — compile-verified
//
#include <hip/hip_runtime.h>
#include <hip/hip_bf16.h>
#include <math.h>

// ---------------- problem constants ----------------
#define B_      256
#define DIN_    2048
#define D_      512
#define C_      1000
#define N_      50000
#define M_      (N_ + B_)     // 50256
#define FK_     100           // FILTER_K

typedef __attribute__((ext_vector_type(2))) float v2f;
typedef __attribute__((ext_vector_type(8))) float v8f;

// D = A(16x4 f32) * B(4x16 f32) + C(16x16 f32), wave32 WMMA
__device__ inline v8f wmma4(v2f a, v2f b, v8f c) {
  return __builtin_amdgcn_wmma_f32_16x16x4_f32(
      /*neg_a=*/false, a, /*neg_b=*/false, b,
      /*c_mod=*/(short)0, c, /*reuse_a=*/false, /*reuse_b=*/false);
}

// ---------------- K1: z = x @ Wf   [256x2048]@[2048x512] ----------------
// One wave computes a 16(M) x 32(N) tile: A frag reused for two B frags.
__global__ void k_gemm_z(const float* __restrict__ x,
                         const float* __restrict__ Wf,
                         float* __restrict__ z) {
  const int lane = threadIdx.x & 31;
  const int w    = blockIdx.x * (blockDim.x >> 5) + (threadIdx.x >> 5);
  const int mt = w >> 4, nt = w & 15;           // 16 x 16 wave tiles
  const int row0 = mt * 16, col0 = nt * 32;
  const int l16 = lane & 15, half = lane >> 4, kh = half * 2;
  v8f c0 = {}; v8f c1 = {};
  const float* xr = x + (size_t)(row0 + l16) * DIN_;
  for (int k = 0; k < DIN_; k += 4) {
    v2f a  = { xr[k + kh], xr[k + kh + 1] };
    const float* b0p = Wf + (size_t)(k + kh) * D_ + col0 + l16;
    v2f b0 = { b0p[0],      b0p[D_] };
    v2f b1 = { b0p[16],     b0p[D_ + 16] };
    c0 = wmma4(a, b0, c0);
    c1 = wmma4(a, b1, c1);
  }
  const int rbase = half * 8;
  for (int r = 0; r < 8; ++r) {
    const int row = row0 + rbase + r;
    z[(size_t)row * D_ + col0 + l16]      = c0[r];
    z[(size_t)row * D_ + col0 + 16 + l16] = c1[r];
  }
}

// ---------------- K2: p = z @ Wc^T + bc   [256x512]@[512x1000] ----------------
__global__ void k_gemm_logits(const float* __restrict__ z,
                              const float* __restrict__ Wc,
                              const float* __restrict__ bc,
                              float* __restrict__ p) {
  const int lane = threadIdx.x & 31;
  const int w    = blockIdx.x * (blockDim.x >> 5) + (threadIdx.x >> 5);
  const int mt = w / 63, nt = w % 63;           // N padded to 1008
  const int row0 = mt * 16, col0 = nt * 16;
  const int l16 = lane & 15, half = lane >> 4, kh = half * 2;
  const int col = col0 + l16;
  const bool colOk = (col < C_);
  const float* zr = z + (size_t)(row0 + l16) * D_;
  const float* wc = colOk ? (Wc + (size_t)col * D_) : Wc;  // guarded below
  v8f c = {};
  for (int k = 0; k < D_; k += 4) {
    v2f a = { zr[k + kh], zr[k + kh + 1] };
    v2f b = { colOk ? wc[k + kh] : 0.f, colOk ? wc[k + kh + 1] : 0.f };
    c = wmma4(a, b, c);
  }
  if (colOk) {
    const float bias = bc[col];
    const int rbase = half * 8;
    for (int r = 0; r < 8; ++r)
      p[(size_t)(row0 + rbase + r) * C_ + col] = c[r] + bias;
  }
}

// ---------------- K3: per-row argmax + softmax entropy (wave per row) ----------------
__global__ void k_row_stats(const float* __restrict__ p,
                            float* __restrict__ ent_new,
                            int* __restrict__ yhat) {
  const int row = blockIdx.x, lane = threadIdx.x;
  const float* pr = p + (size_t)row * C_;
  float best = -3.402823466e38f; int bidx = 0x7fffffff;
  for (int j = lane; j < C_; j += 32) {
    float v = pr[j];
    if (v > best) { best = v; bidx = j; }      // first-index tie within lane
  }
  for (int off = 16; off; off >>= 1) {
    float ov = __shfl_xor(best, off, 32);
    int   oi = __shfl_xor(bidx, off, 32);
    if (ov > best || (ov == best && oi < bidx)) { best = ov; bidx = oi; }
  }
  float s = 0.f, t = 0.f;
  for (int j = lane; j < C_; j += 32) {
    float d = pr[j] - best;
    float e = expf(d);
    s += e; t += d * e;
  }
  for (int off = 16; off; off >>= 1) {
    s += __shfl_xor(s, off, 32);
    t += __shfl_xor(t, off, 32);
  }
  if (lane == 0) { ent_new[row] = logf(s) - t / s; yhat[row] = bidx; }
}

// ---------------- K4: zero / histogram / prefix / scatter ----------------
__global__ void k_zero_i32(int* __restrict__ a, int n) {
  int i = blockIdx.x * blockDim.x + threadIdx.x;
  if (i < n) a[i] = 0;
}

__global__ void k_hist(const int* __restrict__ labels_idx,
                       const int* __restrict__ yhat,
                       int* __restrict__ cnt) {
  int m = blockIdx.x * blockDim.x + threadIdx.x;
  if (m >= M_) return;
  int c = (m < N_) ? labels_idx[m] : yhat[m - N_];
  atomicAdd(&cnt[c], 1);
}

__global__ void k_prefix(const int* __restrict__ cnt, int* __restrict__ offs) {
  if (blockIdx.x == 0 && threadIdx.x == 0) {
    int acc = 0;
    for (int c = 0; c < C_; ++c) { offs[c] = acc; acc += cnt[c]; }
    offs[C_] = acc;
  }
}

__global__ void k_scatter(const int* __restrict__ labels_idx,
                          const int* __restrict__ yhat,
                          const float* __restrict__ ent_in,
                          const float* __restrict__ ent_new,
                          const int* __restrict__ offs,
                          int* __restrict__ cursor,
                          int* __restrict__ bidx, float* __restrict__ bent) {
  int m = blockIdx.x * blockDim.x + threadIdx.x;
  if (m >= M_) return;
  int c; float e;
  if (m < N_) { c = labels_idx[m]; e = ent_in[m]; }
  else        { c = yhat[m - N_];  e = ent_new[m - N_]; }
  int pos = offs[c] + atomicAdd(&cursor[c], 1);
  bidx[pos] = m; bent[pos] = e;
}

// ---------------- K5: per-class rank (stable by (ent, index)) -> compact selected ----
__global__ void k_select(const int* __restrict__ offs,
                         const int* __restrict__ bidx,
                         const float* __restrict__ bent,
                         int* __restrict__ sel_cnt, int* __restrict__ sel) {
  const int c = blockIdx.x;
  const int base = offs[c];
  const int n = offs[c + 1] - base;
  for (int i = threadIdx.x; i < n; i += blockDim.x) {
    bool keep;
    if (n <= FK_) {
      keep = true;
    } else {
      const float ei = bent[base + i];
      const int   mi = bidx[base + i];
      int rank = 0;
      for (int j = 0; j < n; ++j) {
        const float ej = bent[base + j];
        const int   mj = bidx[base + j];
        rank += (ej < ei) || (ej == ei && mj < mi);
      }
      keep = rank < FK_;
    }
    if (keep) {
      int slot = atomicAdd(&sel_cnt[c], 1);
      sel[base + slot] = bidx[base + i];
    }
  }
}

// ---------------- K6: weights[:,c] = normalize(sum of normalized selected rows) ------
// One wave per class. Each lane owns 16 elements of D=512 (stride 32, coalesced).
__global__ void k_weights(const float* __restrict__ supports,
                          const float* __restrict__ z,
                          const int* __restrict__ sel,
                          const int* __restrict__ offs,
                          const int* __restrict__ sel_cnt,
                          float* __restrict__ W) {
  const int c = blockIdx.x;
  const int lane = threadIdx.x;
  float acc[16];
#pragma unroll
  for (int i = 0; i < 16; ++i) acc[i] = 0.f;
  const int n = sel_cnt[c];
  const int base = offs[c];
  for (int s = 0; s < n; ++s) {
    const int m = sel[base + s];
    const float* row = (m < N_) ? supports + (size_t)m * D_
                                : z + (size_t)(m - N_) * D_;
    if (s + 1 < n) {
      const int m2 = sel[base + s + 1];
      const float* nxt = (m2 < N_) ? supports + (size_t)m2 * D_
                                   : z + (size_t)(m2 - N_) * D_;
      __builtin_prefetch(nxt + lane, 0, 0);    // global_prefetch_b8
    }
    float v[16]; float ss = 0.f;
#pragma unroll
    for (int i = 0; i < 16; ++i) { v[i] = row[lane + 32 * i]; ss += v[i] * v[i]; }
    for (int off = 16; off; off >>= 1) ss += __shfl_xor(ss, off, 32);
    const float inv = 1.0f / fmaxf(sqrtf(ss), 1e-12f);
#pragma unroll
    for (int i = 0; i < 16; ++i) acc[i] += v[i] * inv;
  }
  // column norm over D
  float cs = 0.f;
#pragma unroll
  for (int i = 0; i < 16; ++i) cs += acc[i] * acc[i];
  for (int off = 16; off; off >>= 1) cs += __shfl_xor(cs, off, 32);
  const float winv = 1.0f / fmaxf(sqrtf(cs), 1e-12f);
#pragma unroll
  for (int i = 0; i < 16; ++i)
    W[(size_t)(lane + 32 * i) * C_ + c] = acc[i] * winv;
}

// ---------------- K7: out = z @ W   [256x512]@[512x1000] ----------------
__global__ void k_gemm_out(const float* __restrict__ z,
                           const float* __restrict__ W,
                           float* __restrict__ out) {
  const int lane = threadIdx.x & 31;
  const int w    = blockIdx.x * (blockDim.x >> 5) + (threadIdx.x >> 5);
  const int mt = w / 63, nt = w % 63;
  const int row0 = mt * 16, col0 = nt * 16;
  const int l16 = lane & 15, half = lane >> 4, kh = half * 2;
  const int col = col0 + l16;
  const bool colOk = (col < C_);
  const float* zr = z + (size_t)(row0 + l16) * D_;
  v8f c = {};
  for (int k = 0; k < D_; k += 4) {
    v2f a = { zr[k + kh], zr[k + kh + 1] };
    v2f b = { colOk ? W[(size_t)(k + kh) * C_ + col] : 0.f,
              colOk ? W[(size_t)(k + kh + 1) * C_ + col] : 0.f };
    c = wmma4(a, b, c);
  }
  if (colOk) {
    const int rbase = half * 8;
    for (int r = 0; r < 8; ++r)
      out[(size_t)(row0 + rbase + r) * C_ + col] = c[r];
  }
}

// ---------------- launch ----------------
extern "C" void kernel_launch(void* const* d_in, const int* in_sizes, int n_in,
                              void* d_out, int out_size, void* d_ws, size_t ws_size,
                              hipStream_t stream) {
  (void)in_sizes; (void)n_in; (void)out_size; (void)ws_size;
  const float* x      = (const float*)d_in[0];
  const float* Wf     = (const float*)d_in[1];
  const float* Wc     = (const float*)d_in[2];
  const float* bc     = (const float*)d_in[3];
  const float* sup    = (const float*)d_in[4];
  const float* ent_in = (const float*)d_in[5];
  const int*   lidx   = (const int*)d_in[6];
  float* out = (float*)d_out;

  char* ws = (char*)d_ws;
  size_t o = 0;
  auto take = [&](size_t bytes) -> char* {
    char* p = ws + o; o += (bytes + 255) & ~(size_t)255; return p;
  };
  float* z        = (float*)take((size_t)B_ * D_ * 4);      // 512 KB
  float* p        = (float*)take((size_t)B_ * C_ * 4);      // 1000 KB
  float* ent_new  = (float*)take((size_t)B_ * 4);
  int*   yhat     = (int*)  take((size_t)B_ * 4);
  int*   counters = (int*)  take((size_t)3 * C_ * 4);       // cnt | cursor | sel_cnt
  int*   offs     = (int*)  take((size_t)(C_ + 1) * 4);
  int*   bidx     = (int*)  take((size_t)M_ * 4);
  float* bent     = (float*)take((size_t)M_ * 4);
  int*   sel      = (int*)  take((size_t)M_ * 4);
  float* W        = (float*)take((size_t)D_ * C_ * 4);      // 2 MB
  int* cnt = counters, *cursor = counters + C_, *sel_cnt = counters + 2 * C_;

  // 1) z = x @ Wf : 256 waves (16x32 tiles), 8 waves/block
  k_gemm_z<<<32, 256, 0, stream>>>(x, Wf, z);
  // 2) p = z @ Wc^T + bc : 16 x 63 = 1008 wave tiles
  k_gemm_logits<<<126, 256, 0, stream>>>(z, Wc, bc, p);
  // 3) argmax + entropy, wave per row
  k_row_stats<<<B_, 32, 0, stream>>>(p, ent_new, yhat);
  // 4) bucket by class
  k_zero_i32<<<(3 * C_ + 255) / 256, 256, 0, stream>>>(counters, 3 * C_);
  k_hist<<<(M_ + 255) / 256, 256, 0, stream>>>(lidx, yhat, cnt);
  k_prefix<<<1, 32, 0, stream>>>(cnt, offs);
  k_scatter<<<(M_ + 255) / 256, 256, 0, stream>>>(lidx, yhat, ent_in, ent_new,
                                                  offs, cursor, bidx, bent);
  // 5) per-class top-100 by (entropy, index)
  k_select<<<C_, 256, 0, stream>>>(offs, bidx, bent, sel_cnt, sel);
  // 6) weights accumulate + column normalize, wave per class
  k_weights<<<C_, 32, 0, stream>>>(sup, z, sel, offs, sel_cnt, W);
  // 7) out = z @ W
  k_gemm_out<<<126, 256, 0, stream>>>(z, W, out);
}